// QueryAndGroup_22625887715918
// MI455X (gfx1250) — compile-verified
//
#include <hip/hip_runtime.h>

typedef __attribute__((ext_vector_type(2))) float v2f;
typedef __attribute__((ext_vector_type(8))) float v8f;

#define B_  2
#define N_  16384
#define K_  4096
#define C_  128
#define NS  32
#define RAD 0.4f
#define RAD2 (RAD * RAD)

// ---------------------------------------------------------------------------
// Kernel 1: ball query via V_WMMA_F32_16X16X4_F32.
// One wave handles a tile of 16 query points, sweeping N in tiles of 16 points.
// d2[m][n] = |q_m|^2 + |p_n|^2 - 2 q_m . p_n is computed as a single WMMA:
//   A[m][:] = (-2qx, -2qy, -2qz, |q|^2)   (K=4, pads the 3D dot product)
//   B[:][n] = ( px,   py,   pz,  1     )
//   C[m][n] = |p_n|^2
// A-layout (ISA 7.12.2, 32-bit A 16x4): lane L<16 holds K=0,1 of row L;
// lane L+16 holds K=2,3 of row L.  B symmetric (K split across lane halves).
// C/D: VGPR r holds row r (lanes 0-15) and row r+8 (lanes 16-31), col = lane&15.
//
// Selection of the first-32 in-radius indices per row uses wave32 ballot +
// popcount prefix; per-row counts mirrored in lanes (row = lane & 15).
// Fast paths (both uniform branches, EXEC stays all-ones around the WMMA):
//   - all 16 rows full  -> break (dense central queries finish after ~2K pts)
//   - tile has no hits  -> skip the 8-row selection pass entirely (dominant
//     case for outer queries that must scan all 16384 points)
// ---------------------------------------------------------------------------
__global__ __launch_bounds__(256) void ball_query_wmma(
    const float* __restrict__ xyz,      // (B, N, 3)
    const float* __restrict__ new_xyz,  // (B, K, 3)
    int* __restrict__ idx_out)          // (B, K, NS)
{
    const int lane = threadIdx.x & 31;
    const int wave = threadIdx.x >> 5;
    const int w    = blockIdx.x * 8 + wave;            // 512 waves total
    const int tilesPerB = K_ / 16;
    const int b     = w / tilesPerB;
    const int qbase = (w % tilesPerB) * 16;

    const float* xb = xyz     + (size_t)b * N_ * 3;
    const float* qb = new_xyz + (size_t)b * K_ * 3;

    const int mrow = lane & 15;

    // Load this wave's 16 query points; build A operand per ISA layout.
    const float* Q = qb + (size_t)(qbase + mrow) * 3;
    const float qx = Q[0], qy = Q[1], qz = Q[2];
    const float qn2 = qx * qx + qy * qy + qz * qz;
    v2f av;
    if (lane < 16) { av[0] = -2.0f * qx; av[1] = -2.0f * qy; }
    else           { av[0] = -2.0f * qz; av[1] = qn2;        }

    int cnt = 0;  // hits for row (lane & 15), mirrored in both lane halves

    for (int pbase = 0; pbase < N_; pbase += 16) {
        // Uniform early exit once every row has >= NS hits.
        const unsigned full = __builtin_amdgcn_ballot_w32(cnt >= NS);
        if (full == 0xFFFFFFFFu) break;

        // Load 16 points; build B operand and C = |p_n|^2 broadcast.
        const float* P = xb + (size_t)(pbase + mrow) * 3;
        const float px = P[0], py = P[1], pz = P[2];
        const float pn2 = px * px + py * py + pz * pz;
        v2f bv;
        if (lane < 16) { bv[0] = px; bv[1] = py;   }
        else           { bv[0] = pz; bv[1] = 1.0f; }
        v8f cv;
#pragma unroll
        for (int i = 0; i < 8; ++i) cv[i] = pn2;

        // D = A x B + C  ->  16x16 tile of squared distances.
        v8f d2 = __builtin_amdgcn_wmma_f32_16x16x4_f32(
            false, av, false, bv, (short)0, cv, false, false);

        // Empty-tile fast path: min-reduce the 8 rows held by this lane,
        // one ballot over the wave; skip selection if the tile has no hits.
        float dmin = d2[0];
#pragma unroll
        for (int r = 1; r < 8; ++r) dmin = fminf(dmin, d2[r]);
        const unsigned any = __builtin_amdgcn_ballot_w32(dmin < RAD2);
        if (any == 0u) continue;

#pragma unroll
        for (int r = 0; r < 8; ++r) {
            const bool pred = d2[r] < RAD2;
            const unsigned mask = __builtin_amdgcn_ballot_w32(pred);
            // bits [15:0] = row r, bits [31:16] = row r+8
            const int cnt_lo = __builtin_amdgcn_readlane(cnt, r);
            const int cnt_hi = __builtin_amdgcn_readlane(cnt, r + 8);
            const int base = (lane < 16) ? cnt_lo : cnt_hi;

            unsigned prior = mask & ((1u << lane) - 1u);
            if (lane >= 16) prior &= 0xFFFF0000u;
            const int pos = base + __popc(prior);

            if (pred && pos < NS) {
                const int row = (lane < 16) ? r : (r + 8);
                const size_t ob = ((size_t)b * K_ + qbase + row) * NS;
                idx_out[ob + pos] = pbase + mrow;
            }
            const int add_lo = __popc(mask & 0xFFFFu);
            const int add_hi = __popc(mask >> 16);
            if (mrow == r)     cnt += add_lo;
            if (mrow == r + 8) cnt += add_hi;
        }
    }

    // Pad unfilled slots with first hit (or 0 if none) — lanes 0..15, one row each.
    if (lane < 16) {
        const size_t ob = ((size_t)b * K_ + qbase + mrow) * NS;
        const int c = (cnt < NS) ? cnt : NS;
        const int first = (c > 0) ? idx_out[ob] : 0;
        for (int j = c; j < NS; ++j) idx_out[ob + j] = first;
    }
}

// ---------------------------------------------------------------------------
// Kernel 2: gather grouped_xyz (centered, /RADIUS) and grouped_features.
// One 256-thread block per (b, k). features (16.8 MB) is L2-resident on the
// 192 MB L2; output writes are 128 B coalesced bursts -> HBM-write bound
// (~137 MB of writes -> ~6 us floor at 23.3 TB/s).
// ---------------------------------------------------------------------------
__global__ __launch_bounds__(256) void group_gather(
    const float* __restrict__ xyz,      // (B, N, 3)
    const float* __restrict__ new_xyz,  // (B, K, 3)
    const float* __restrict__ feat,     // (B, C, N)
    const int*   __restrict__ idx,      // (B, K, NS)
    float* __restrict__ out_xyz,        // (B, 3, K, NS)
    float* __restrict__ out_feat)       // (B, C, K, NS)
{
    __shared__ int sidx[NS];
    const int bk = blockIdx.x;
    const int b  = bk / K_;
    const int k  = bk % K_;
    const int t  = threadIdx.x;

    if (t < NS) sidx[t] = idx[((size_t)b * K_ + k) * NS + t];
    __syncthreads();

    const int s = t & 31;
    const int j = sidx[s];

    if (t < 96) {
        const int d = t >> 5;
        const float v = (xyz[((size_t)b * N_ + j) * 3 + d] -
                         new_xyz[((size_t)b * K_ + k) * 3 + d]) * (1.0f / RAD);
        out_xyz[(((size_t)b * 3 + d) * K_ + k) * NS + s] = v;
    }

    const float* fb = feat     + (size_t)b * C_ * N_;
    float*       ob = out_feat + (size_t)b * C_ * K_ * NS;
#pragma unroll 4
    for (int c = t >> 5; c < C_; c += 8) {
        ob[((size_t)c * K_ + k) * NS + s] = fb[(size_t)c * N_ + j];
    }
}

extern "C" void kernel_launch(void* const* d_in, const int* in_sizes, int n_in,
                              void* d_out, int out_size, void* d_ws, size_t ws_size,
                              hipStream_t stream) {
    (void)in_sizes; (void)n_in; (void)out_size; (void)ws_size;

    const float* xyz     = (const float*)d_in[0];  // B*N*3
    const float* new_xyz = (const float*)d_in[1];  // B*K*3
    const float* feat    = (const float*)d_in[2];  // B*C*N

    int*   idx_ws   = (int*)d_ws;                          // B*K*NS ints = 1 MB
    float* out_xyz  = (float*)d_out;                       // B*3*K*NS
    float* out_feat = out_xyz + (size_t)B_ * 3 * K_ * NS;  // B*C*K*NS

    const int waves = B_ * (K_ / 16);  // 512 waves, 8 per block
    ball_query_wmma<<<waves / 8, 256, 0, stream>>>(xyz, new_xyz, idx_ws);
    group_gather<<<B_ * K_, 256, 0, stream>>>(xyz, new_xyz, feat, idx_ws,
                                              out_xyz, out_feat);
}